// ReadoutHead_63797444215135
// MI455X (gfx1250) — compile-verified
//
#include <hip/hip_runtime.h>
#include <math.h>

typedef __attribute__((ext_vector_type(2))) float v2f;
typedef __attribute__((ext_vector_type(8))) float v8f;

#define POOLED 11520
#define HALF   5760
#define HIDDEN 11520
#define NCLS   1000
#define BATCH  32

// GEMM1 tiling: grid (90 N-tiles x KSPLIT1 K-splits); per block K-range 1440.
#define KSPLIT1 8
#define KRANGE1 (POOLED / KSPLIT1)      // 1440
#define KC      96                      // LDS chunk; 1440 = 15 * 96
#define XS_STRIDE 100                   // 100 mod 64 = 36 -> conflict-free b64 A-frags

// GEMM2 tiling: grid (63 x KSPLIT2); 8 waves/block -> 360 K per wave.
#define KSPLIT2 4
#define NPAD    1008                    // 63 * 16

// ---------------------------------------------------------------------------
// Kernel 1: per-(b,c) plane pooling: mean, max, (mean |x|^3)^(1/3)
// ---------------------------------------------------------------------------
__global__ __launch_bounds__(256) void pool_kernel(
    const float* __restrict__ x, float* __restrict__ pooled,
    int C, int HW, int base)
{
    __shared__ float s_sum[256], s_max[256], s_p[256];
    const int plane = blockIdx.x;            // b*C + c
    const int b = plane / C, c = plane % C;
    const float* p = x + (size_t)plane * HW;

    float sum = 0.f, mx = -INFINITY, ps = 0.f;
    for (int i = threadIdx.x; i < HW; i += 256) {
        float v = p[i];
        sum += v;
        mx = fmaxf(mx, v);
        float a = fmaxf(fabsf(v), 1e-6f);
        ps += a * a * a;
    }
    s_sum[threadIdx.x] = sum; s_max[threadIdx.x] = mx; s_p[threadIdx.x] = ps;
    __syncthreads();
    for (int off = 128; off > 0; off >>= 1) {
        if ((int)threadIdx.x < off) {
            s_sum[threadIdx.x] += s_sum[threadIdx.x + off];
            s_max[threadIdx.x] = fmaxf(s_max[threadIdx.x], s_max[threadIdx.x + off]);
            s_p[threadIdx.x]  += s_p[threadIdx.x + off];
        }
        __syncthreads();
    }
    if (threadIdx.x == 0) {
        float inv = 1.f / (float)HW;
        float* row = pooled + (size_t)b * POOLED + base;
        row[c]         = s_sum[0] * inv;
        row[C + c]     = s_max[0];
        row[2 * C + c] = cbrtf(s_p[0] * inv);
    }
}

// ---------------------------------------------------------------------------
// Kernel 2: butterfly mix + RMS norm, in-place on X (one block per batch row)
// ---------------------------------------------------------------------------
__global__ __launch_bounds__(256) void butterfly_rms_kernel(
    float* __restrict__ X, const float* __restrict__ theta,
    const float* __restrict__ rms_w)
{
    __shared__ float row[POOLED];    // 46 KB
    __shared__ float s_red[256];
    __shared__ float s_rms;
    float* xr = X + (size_t)blockIdx.x * POOLED;

    float ss = 0.f;
    for (int j = threadIdx.x; j < HALF; j += 256) {
        float a = xr[j], b = xr[j + HALF];
        float t = theta[j];
        float s, c;
        sincosf(t, &s, &c);
        float u = c * a - s * b;
        float v = s * a + c * b;
        row[j] = u; row[j + HALF] = v;
        ss += u * u + v * v;
    }
    s_red[threadIdx.x] = ss;
    __syncthreads();
    for (int off = 128; off > 0; off >>= 1) {
        if ((int)threadIdx.x < off) s_red[threadIdx.x] += s_red[threadIdx.x + off];
        __syncthreads();
    }
    if (threadIdx.x == 0) s_rms = rsqrtf(s_red[0] / (float)POOLED + 1e-6f);
    __syncthreads();
    float rms = s_rms;
    for (int i = threadIdx.x; i < POOLED; i += 256)
        xr[i] = row[i] * rms * rms_w[i];
}

// ---------------------------------------------------------------------------
// Kernel 3a: GEMM1 partials. Block (bx, by): N columns [bx*128, +128),
// K range [by*1440, +1440). 8 waves; each wave a 16-wide N tile, both
// 16-row M halves. X chunks staged in LDS; w1 streamed from HBM.
// Writes fp32 partial C tiles (no bias/gelu) to P1[by][m][n].
// ---------------------------------------------------------------------------
__global__ __launch_bounds__(256) void gemm1_kernel(
    const float* __restrict__ X, const float* __restrict__ w1,
    float* __restrict__ P1)
{
    __shared__ float Xs[32 * XS_STRIDE];     // 12.8 KB
    const int t    = threadIdx.x;
    const int wave = t >> 5;
    const int lane = t & 31;
    const int h    = lane >> 4;     // K-half selector (ISA A/B lane layout)
    const int nl   = lane & 15;
    const int n    = blockIdx.x * 128 + wave * 16 + nl;
    const int kbase = blockIdx.y * KRANGE1;

    v8f acc0 = {};   // rows 0..15
    v8f acc1 = {};   // rows 16..31

    for (int k0 = kbase; k0 < kbase + KRANGE1; k0 += KC) {
        // Stage X[0:32, k0:k0+96] into LDS: 768 float4s, 3 reps of 256 threads
        #pragma unroll
        for (int rep = 0; rep < 3; ++rep) {
            int lin = t + rep * 256;
            int m   = lin / 24;               // 24 float4 per row
            int c4  = (lin % 24) << 2;
            float4 v = *(const float4*)&X[(size_t)m * POOLED + k0 + c4];
            *(float4*)&Xs[m * XS_STRIDE + c4] = v;
        }
        __syncthreads();

        const float* wcol = w1 + (size_t)k0 * HIDDEN + n;
        #pragma unroll 4
        for (int kk = 0; kk < KC; kk += 4) {
            v2f a0 = *(const v2f*)&Xs[nl * XS_STRIDE + kk + 2 * h];
            v2f a1 = *(const v2f*)&Xs[(16 + nl) * XS_STRIDE + kk + 2 * h];
            v2f bf;
            bf.x = wcol[(size_t)(kk + 2 * h) * HIDDEN];
            bf.y = wcol[(size_t)(kk + 2 * h + 1) * HIDDEN];
            acc0 = __builtin_amdgcn_wmma_f32_16x16x4_f32(
                false, a0, false, bf, (short)0, acc0, false, false);
            acc1 = __builtin_amdgcn_wmma_f32_16x16x4_f32(
                false, a1, false, bf, (short)0, acc1, false, false);
        }
        __syncthreads();
    }

    float* Pслab = P1 + (size_t)blockIdx.y * 32 * HIDDEN;
    #pragma unroll
    for (int g = 0; g < 8; ++g) {
        Pслab[(size_t)(g + 8 * h) * HIDDEN + n]      = acc0[g];
        Pслab[(size_t)(16 + g + 8 * h) * HIDDEN + n] = acc1[g];
    }
}

// ---------------------------------------------------------------------------
// Kernel 3b: reduce KSPLIT1 partials + bias + exact GELU -> H1 (32 x 11520)
// ---------------------------------------------------------------------------
__global__ __launch_bounds__(256) void gelu_reduce_kernel(
    const float* __restrict__ P1, const float* __restrict__ b1,
    float* __restrict__ H1)
{
    int idx = blockIdx.x * 256 + threadIdx.x;      // 0 .. 32*11520-1
    int ncol = idx % HIDDEN;
    float v = b1[ncol];
    #pragma unroll
    for (int y = 0; y < KSPLIT1; ++y)
        v += P1[(size_t)y * 32 * HIDDEN + idx];
    v = 0.5f * v * (1.f + erff(v * 0.70710678118654752f));
    H1[idx] = v;
}

// ---------------------------------------------------------------------------
// Kernel 4a: GEMM2 partials. Block (bx, by): N tile [bx*16,+16),
// K range [by*2880, +2880) split over 8 waves (360 each). Wave partials
// reduced through LDS; block partial stored to P2[by][m][ncol].
// ---------------------------------------------------------------------------
__global__ __launch_bounds__(256) void gemm2_kernel(
    const float* __restrict__ H1, const float* __restrict__ w2,
    float* __restrict__ P2)
{
    __shared__ float red[8 * 32 * 16];   // 16 KB of wave partials
    const int t    = threadIdx.x;
    const int wave = t >> 5;
    const int lane = t & 31;
    const int h    = lane >> 4;
    const int nl   = lane & 15;
    const int n    = blockIdx.x * 16 + nl;
    const int kbeg = blockIdx.y * (POOLED / KSPLIT2) + wave * (POOLED / KSPLIT2 / 8);
    const int kend = kbeg + (POOLED / KSPLIT2 / 8);   // 360 K per wave

    v8f acc0 = {};
    v8f acc1 = {};
    for (int k = kbeg; k < kend; k += 4) {
        int kk = k + 2 * h;
        v2f a0 = *(const v2f*)&H1[(size_t)nl * HIDDEN + kk];
        v2f a1 = *(const v2f*)&H1[(size_t)(16 + nl) * HIDDEN + kk];
        v2f bf;
        bf.x = (n < NCLS) ? w2[(size_t)kk * NCLS + n] : 0.f;
        bf.y = (n < NCLS) ? w2[(size_t)(kk + 1) * NCLS + n] : 0.f;
        acc0 = __builtin_amdgcn_wmma_f32_16x16x4_f32(
            false, a0, false, bf, (short)0, acc0, false, false);
        acc1 = __builtin_amdgcn_wmma_f32_16x16x4_f32(
            false, a1, false, bf, (short)0, acc1, false, false);
    }

    float* myred = &red[(wave * 32 + lane) * 16];
    #pragma unroll
    for (int g = 0; g < 8; ++g) { myred[g] = acc0[g]; myred[8 + g] = acc1[g]; }
    __syncthreads();

    // 512 tile elements (32 M x 16 N), 2 per thread
    for (int idx = t; idx < 512; idx += 256) {
        int m  = idx >> 4, nn = idx & 15;
        int srclane = (((m & 15) >> 3) << 4) + nn;   // h*16 + nl
        int slot    = (m & 7) + ((m >> 4) ? 8 : 0);  // acc0 vs acc1 half
        float v = 0.f;
        #pragma unroll
        for (int w = 0; w < 8; ++w) v += red[(w * 32 + srclane) * 16 + slot];
        int ncol = blockIdx.x * 16 + nn;             // < NPAD always
        P2[((size_t)blockIdx.y * 32 + m) * NPAD + ncol] = v;
    }
}

// ---------------------------------------------------------------------------
// Kernel 4b: reduce KSPLIT2 partials + b2 -> out (32 x 1000)
// ---------------------------------------------------------------------------
__global__ __launch_bounds__(256) void out_reduce_kernel(
    const float* __restrict__ P2, const float* __restrict__ b2,
    float* __restrict__ out)
{
    int idx = blockIdx.x * 256 + threadIdx.x;      // 0 .. 32*1000-1
    if (idx >= BATCH * NCLS) return;
    int m = idx / NCLS, ncol = idx % NCLS;
    float v = b2[ncol];
    #pragma unroll
    for (int y = 0; y < KSPLIT2; ++y)
        v += P2[((size_t)y * 32 + m) * NPAD + ncol];
    out[idx] = v;
}

// ---------------------------------------------------------------------------
extern "C" void kernel_launch(void* const* d_in, const int* in_sizes, int n_in,
                              void* d_out, int out_size, void* d_ws, size_t ws_size,
                              hipStream_t stream) {
    const float* f0    = (const float*)d_in[0];
    const float* f1    = (const float*)d_in[1];
    const float* f2    = (const float*)d_in[2];
    const float* f3    = (const float*)d_in[3];
    const float* theta = (const float*)d_in[4];
    const float* rms_w = (const float*)d_in[5];
    const float* w1    = (const float*)d_in[6];
    const float* b1    = (const float*)d_in[7];
    const float* w2    = (const float*)d_in[8];
    const float* b2    = (const float*)d_in[9];
    float* outp = (float*)d_out;

    float* X  = (float*)d_ws;                         // 32 x 11520
    float* H1 = X  + (size_t)BATCH * POOLED;          // 32 x 11520
    float* P1 = H1 + (size_t)BATCH * HIDDEN;          // KSPLIT1 x 32 x 11520
    float* P2 = P1 + (size_t)KSPLIT1 * BATCH * HIDDEN;// KSPLIT2 x 32 x 1008

    // Pooling: base offsets 0 / 768 / 2304 / 5376 (3*C per scale)
    pool_kernel<<<BATCH * 256,  256, 0, stream>>>(f0, X,  256, 4096, 0);
    pool_kernel<<<BATCH * 512,  256, 0, stream>>>(f1, X,  512, 1024, 768);
    pool_kernel<<<BATCH * 1024, 256, 0, stream>>>(f2, X, 1024,  256, 2304);
    pool_kernel<<<BATCH * 2048, 256, 0, stream>>>(f3, X, 2048,   64, 5376);

    butterfly_rms_kernel<<<BATCH, 256, 0, stream>>>(X, theta, rms_w);

    gemm1_kernel<<<dim3(HIDDEN / 128, KSPLIT1), 256, 0, stream>>>(X, w1, P1);   // 720 blocks
    gelu_reduce_kernel<<<(BATCH * HIDDEN) / 256, 256, 0, stream>>>(P1, b1, H1); // 1440 blocks

    gemm2_kernel<<<dim3(NPAD / 16, KSPLIT2), 256, 0, stream>>>(H1, w2, P2);     // 252 blocks
    out_reduce_kernel<<<(BATCH * NCLS + 255) / 256, 256, 0, stream>>>(P2, b2, outp);
}